// Hippocampus_48017734369418
// MI455X (gfx1250) — compile-verified
//
#include <hip/hip_runtime.h>
#include <hip/hip_bf16.h>
#include <math.h>

// ---------------- problem constants (match reference) ----------------
#define NQ      128          // B
#define ND      256          // D
#define TOPK    32
#define INV_TAU 10.0f        // 1 / 0.1
#define BETA    0.5f
#define SIGMA   1.0f
#define NEG_INF (-3.402823466e38f)

// ---------------- tiling ----------------
#define NBLK1   128          // phase-1 blocks (column partitions)
#define CHUNK   32           // K columns staged per chunk
#define CPAD    260          // padded floats per K-column in LDS (bank spread)

typedef __attribute__((ext_vector_type(2))) float v2f;
typedef __attribute__((ext_vector_type(4))) float v4f;
typedef __attribute__((ext_vector_type(8))) float v8f;
typedef __attribute__((ext_vector_type(4))) int   v4i;

#if __has_builtin(__builtin_amdgcn_global_load_async_to_lds_b128) && \
    __has_builtin(__builtin_amdgcn_s_wait_asynccnt)
#define USE_ASYNC 1
#else
#define USE_ASYNC 0
#endif

// =====================================================================
// Kernel 0: L2-normalize queries (matches F.normalize with eps=1e-12)
// =====================================================================
__global__ __launch_bounds__(256) void hippo_normalize(
    const float* __restrict__ query, float* __restrict__ qn)
{
    __shared__ float red[256];
    const int row = blockIdx.x;
    const int tid = threadIdx.x;
    float x = query[row * ND + tid];
    red[tid] = x * x;
    __syncthreads();
    for (int s = 128; s > 0; s >>= 1) {
        if (tid < s) red[tid] += red[tid + s];
        __syncthreads();
    }
    float norm  = sqrtf(red[0]);
    float scale = 1.0f / fmaxf(norm, 1e-12f);
    qn[row * ND + tid] = x * scale;
}

// ---------------- K-chunk staging: global -> LDS ----------------
__device__ __forceinline__ void stage_async(const float* __restrict__ Kbank,
                                            float* ldsBase, int c)
{
#if USE_ASYNC
    const int tid = threadIdx.x;
    const float* src0 = Kbank + (size_t)c * CHUNK * ND;   // chunk is contiguous
#pragma unroll
    for (int j = 0; j < 8; ++j) {
        int l   = tid + j * 256;       // float4 index 0..2047
        int col = l >> 6;              // 64 float4 per K-row
        int q4  = l & 63;
        const float* g = src0 + col * ND + q4 * 4;
        float* lp      = ldsBase + col * CPAD + q4 * 4;
        __builtin_amdgcn_global_load_async_to_lds_b128(
            (v4i*)const_cast<float*>(g), (v4i*)lp, 0, 0);
    }
#else
    (void)Kbank; (void)ldsBase; (void)c;
#endif
}

__device__ __forceinline__ void stage_sync(const float* __restrict__ Kbank,
                                           float* ldsBase, int c)
{
    const int tid = threadIdx.x;
    const float* src0 = Kbank + (size_t)c * CHUNK * ND;
#pragma unroll
    for (int j = 0; j < 8; ++j) {
        int l   = tid + j * 256;
        int col = l >> 6;
        int q4  = l & 63;
        v4f val = *(const v4f*)(src0 + col * ND + q4 * 4);
        *(v4f*)(ldsBase + col * CPAD + q4 * 4) = val;
    }
}

// =====================================================================
// Kernel 1: streaming f32 WMMA GEMM (q_norm @ K^T / tau - penalty)
//           + per-row streaming top-32 over this block's column range.
// 8 waves / block; wave w owns query rows [16w, 16w+16).
// =====================================================================
__global__ __launch_bounds__(256) void hippo_phase1(
    const float* __restrict__ qn,
    const float* __restrict__ Kbank,
    const float* __restrict__ times,
    const float* __restrict__ qtime,
    float* __restrict__ candv,
    int*   __restrict__ candi,
    int N)
{
    __shared__ float Klds[2 * CHUNK * CPAD];   // ~65 KB double buffer
    __shared__ float topv[NQ * TOPK];          // 16 KB
    __shared__ int   topi[NQ * TOPK];          // 16 KB
    __shared__ float topmin[NQ];               // cached per-row list min

    const int tid   = threadIdx.x;
    const int lane  = tid & 31;
    const int wv    = tid >> 5;
    const int qbase = wv * 16;
    const int m16   = lane & 15;
    const int koff  = (lane >> 4) << 1;        // 0 for lanes 0-15, 2 for 16-31
    const int blk   = blockIdx.x;

    for (int i = tid; i < NQ * TOPK; i += 256) { topv[i] = NEG_INF; topi[i] = 0; }
    for (int i = tid; i < NQ; i += 256) topmin[i] = NEG_INF;

    // A fragments for this wave's 16x256 q-tile, ISA 16x4 f32 layout:
    // lane holds (m = L%16, k = 4*ks + 2*(L/16) + {0,1})
    v2f qreg[64];
#pragma unroll
    for (int ks = 0; ks < 64; ++ks) {
        const float* p = qn + (qbase + m16) * ND + 4 * ks + koff;
        qreg[ks] = *(const v2f*)p;
    }
    const float qt = qtime[0];

    const int chunks_total = N / CHUNK;                        // 15625
    const int per_blk = (chunks_total + NBLK1 - 1) / NBLK1;
    const int cfirst  = blk * per_blk;
    const int cend    = min(cfirst + per_blk, chunks_total);

    __syncthreads();

#if USE_ASYNC
    if (cfirst < cend) stage_async(Kbank, &Klds[0], cfirst);
#endif

    for (int ci = cfirst; ci < cend; ++ci) {
        const int cur = (ci - cfirst) & 1;
        float* curbuf = &Klds[cur * CHUNK * CPAD];
#if USE_ASYNC
        if (ci + 1 < cend) {
            stage_async(Kbank, &Klds[(cur ^ 1) * CHUNK * CPAD], ci + 1);
            __builtin_amdgcn_s_wait_asynccnt(8);   // chunk ci's 8 loads done
        } else {
            __builtin_amdgcn_s_wait_asynccnt(0);
        }
        __syncthreads();
#else
        __syncthreads();
        stage_sync(Kbank, curbuf, ci);
        __syncthreads();
#endif
        // ---- GEMM: 16 rows x 32 cols, K=256, via v_wmma_f32_16x16x4_f32 ----
        v8f acc0 = {};
        v8f acc1 = {};
#pragma unroll
        for (int ks = 0; ks < 64; ++ks) {
            v2f a  = qreg[ks];
            // B frag (4x16): lane holds (k = 4*ks + koff + {0,1}, n = L%16)
            v2f b0 = *(const v2f*)(curbuf + m16 * CPAD + 4 * ks + koff);
            v2f b1 = *(const v2f*)(curbuf + (m16 + 16) * CPAD + 4 * ks + koff);
            acc0 = __builtin_amdgcn_wmma_f32_16x16x4_f32(
                       false, a, false, b0, (short)0, acc0, false, false);
            acc1 = __builtin_amdgcn_wmma_f32_16x16x4_f32(
                       false, a, false, b1, (short)0, acc1, false, false);
        }

        // ---- penalty + streaming exact top-32 insertion ----
        const int c0 = ci * CHUNK;
#pragma unroll
        for (int t = 0; t < 2; ++t) {
            const int   col = c0 + 16 * t + m16;   // lane's column
            const float dt  = qt - times[col];
            const float pen = (BETA / (2.0f * SIGMA * SIGMA)) * dt * dt;
            const int   rowL = qbase + ((lane >> 4) << 3);
#pragma unroll
            for (int r = 0; r < 8; ++r) {
                // C/D layout: VGPR r, lane L -> row r + 8*(L/16), col L%16
                float vv = ((t == 0) ? acc0[r] : acc1[r]) * INV_TAU - pen;
                const int myrow = rowL + r;
                float thresh = topmin[myrow];      // monotone -> safe if stale
                unsigned long long mball = __ballot(vv > thresh);
                while (mball) {
                    int   src  = __ffsll((long long)mball) - 1;
                    float bv   = __shfl(vv, src);
                    int   brow = qbase + r + ((src >> 4) << 3);
                    int   bcol = c0 + 16 * t + (src & 15);
                    // wave-cooperative argmin over the row's 32-entry list
                    float sv = topv[brow * TOPK + lane];
                    float mv = sv; int ms = lane;
#pragma unroll
                    for (int off = 16; off >= 1; off >>= 1) {
                        float ov = __shfl_xor(mv, off);
                        int   os = __shfl_xor(ms, off);
                        if (ov < mv || (ov == mv && os < ms)) { mv = ov; ms = os; }
                    }
                    if (bv > mv) {                 // exact re-check
                        if (lane == ms) {
                            topv[brow * TOPK + ms] = bv;
                            topi[brow * TOPK + ms] = bcol;
                        }
                        float nv = (lane == ms) ? bv : sv;  // new list min
#pragma unroll
                        for (int off = 16; off >= 1; off >>= 1)
                            nv = fminf(nv, __shfl_xor(nv, off));
                        if (lane == 0) topmin[brow] = nv;
                    }
                    mball &= (mball - 1);
                }
            }
        }
        __syncthreads();   // buffer reuse fence for next chunk's prefetch
    }

    __syncthreads();
    // ---- emit this block's 32 candidates per row ----
    const int stride = NBLK1 * TOPK;
#pragma unroll
    for (int r = 0; r < 16; ++r) {
        const int row = qbase + r;
        candv[(size_t)row * stride + blk * TOPK + lane] = topv[row * TOPK + lane];
        candi[(size_t)row * stride + blk * TOPK + lane] = topi[row * TOPK + lane];
    }
}

// =====================================================================
// Kernel 2: per-row merge of 4096 candidates -> exact top-32,
//           softmax, gather V, weighted sum.
// =====================================================================
__global__ __launch_bounds__(256) void hippo_phase2(
    const float* __restrict__ candv, const int* __restrict__ candi,
    const float* __restrict__ Vbank, float* __restrict__ out)
{
    __shared__ float cv[NBLK1 * TOPK];   // 16 KB
    __shared__ int   cidx[NBLK1 * TOPK]; // 16 KB
    __shared__ float rv[256];
    __shared__ int   ri[256];
    __shared__ float selv[TOPK];
    __shared__ int   seli[TOPK];
    __shared__ float attn[TOPK];

    const int row = blockIdx.x;
    const int tid = threadIdx.x;
    const int M   = NBLK1 * TOPK;

    for (int i = tid; i < M; i += 256) {
        cv[i]   = candv[(size_t)row * M + i];
        cidx[i] = candi[(size_t)row * M + i];
    }
    __syncthreads();

    for (int k = 0; k < TOPK; ++k) {
        float bm = NEG_INF; int bi = -1;
        for (int i = tid; i < M; i += 256) {
            float v = cv[i];
            if (v > bm) { bm = v; bi = i; }
        }
        rv[tid] = bm; ri[tid] = bi;
        __syncthreads();
        for (int s = 128; s > 0; s >>= 1) {
            if (tid < s && rv[tid + s] > rv[tid]) {
                rv[tid] = rv[tid + s]; ri[tid] = ri[tid + s];
            }
            __syncthreads();
        }
        if (tid == 0) {
            int w = ri[0];
            selv[k] = rv[0];
            seli[k] = (w >= 0) ? cidx[w] : 0;
            if (w >= 0) cv[w] = NEG_INF;
        }
        __syncthreads();
    }

    if (tid < 32) {    // softmax over the 32 selected logits (wave 0)
        float v  = selv[tid];
        float mx = v;
        for (int off = 16; off >= 1; off >>= 1) mx = fmaxf(mx, __shfl_xor(mx, off));
        float e = __expf(v - mx);
        float s = e;
        for (int off = 16; off >= 1; off >>= 1) s += __shfl_xor(s, off);
        attn[tid] = e / s;
    }
    __syncthreads();

    float acc = 0.0f;
    const int d = tid;                 // 256 threads == 256 dims
    for (int k = 0; k < TOPK; ++k)
        acc += attn[k] * Vbank[(size_t)seli[k] * ND + d];
    out[row * ND + d] = acc;
}

// =====================================================================
extern "C" void kernel_launch(void* const* d_in, const int* in_sizes, int n_in,
                              void* d_out, int out_size, void* d_ws, size_t ws_size,
                              hipStream_t stream)
{
    const float* query = (const float*)d_in[0];
    const float* Kbank = (const float*)d_in[1];
    const float* Vbank = (const float*)d_in[2];
    const float* times = (const float*)d_in[3];
    const float* qtime = (const float*)d_in[4];
    float* out = (float*)d_out;
    const int N = in_sizes[3];

    char*  ws    = (char*)d_ws;
    float* qn    = (float*)ws;                                   // 128 KB
    float* candv = (float*)(ws + 128 * 1024);                    // 2 MB
    int*   candi = (int*)  (ws + 128 * 1024 +
                            (size_t)NQ * NBLK1 * TOPK * sizeof(float));

    hippo_normalize<<<dim3(NQ),    dim3(256), 0, stream>>>(query, qn);
    hippo_phase1   <<<dim3(NBLK1), dim3(256), 0, stream>>>(qn, Kbank, times, qtime,
                                                           candv, candi, N);
    hippo_phase2   <<<dim3(NQ),    dim3(256), 0, stream>>>(candv, candi, Vbank, out);
}